// LOUPESampler_85822036508952
// MI455X (gfx1250) — compile-verified
//
#include <hip/hip_runtime.h>
#include <stdint.h>

// ---------------------------------------------------------------------------
// LOUPE sampler for MI455X (gfx1250, wave32).
//
// Stage 2 (hot, defined FIRST so the disasm snippet shows it):
//   masked_kspace = mask * kspace : 210 MB read, 315 MB write, ~52 MFLOP
//   => pure HBM stream, roofline floor ~524MB / 23.3TB/s ~= 22.5us.
//   b128 non-temporal loads/stores (writes exceed 192MB L2, never re-read),
//   global_prefetch_b8 ahead on the read stream, 3D launch so there is
//   zero integer div/mod in the address math.
//
// Stage 1 (tiny): 320-line prob-mask -> rescale -> threefry rejection-
//   sampling binarize -> preselect endpoints. One 320-thread workgroup,
//   deterministic ordered reductions (no float atomics).
// ---------------------------------------------------------------------------

#define NL   320          // number of k-space lines
#define NB   256          // batch
#define HW   (320 * 320)  // one channel plane
#define SLOPE 5.0f

typedef float v4f __attribute__((ext_vector_type(4)));

// ------------------------ stage 2: apply kernel ----------------------------
// block = (80, 4): threadIdx.x = float4 index along W, threadIdx.y = row-in-
// block; grid = (80, 256): blockIdx.x = row group, blockIdx.y = batch.
// Each thread: prefetch + 2 b128 NT loads + 2 mul + 3 b128 NT stores.
__global__ __launch_bounds__(320)
void loupe_apply_kernel(const float *__restrict__ kspace,
                        const float *__restrict__ mask,
                        float *__restrict__ out_masked,
                        float *__restrict__ out_mask) {
    const int w = threadIdx.x * 4;                    // 0..316 step 4
    const int h = blockIdx.x * 4 + threadIdx.y;       // 0..319
    const int b = blockIdx.y;                         // 0..255

    const float m = mask[h];                          // tiny, cache-resident
    const long base0 = ((long)(b * 2 + 0) * 320 + h) * 320 + w;
    const long base1 = base0 + HW;
    const long mbase = ((long)b * 320 + h) * 320 + w;

    // gfx1250 global_prefetch_b8: pull the read stream ~8KB ahead into GL2.
    __builtin_prefetch(kspace + base0 + 2048, 0, 1);

    const v4f k0 = __builtin_nontemporal_load((const v4f *)(kspace + base0));
    const v4f k1 = __builtin_nontemporal_load((const v4f *)(kspace + base1));
    const v4f m4 = {m, m, m, m};

    __builtin_nontemporal_store(k0 * m, (v4f *)(out_masked + base0));
    __builtin_nontemporal_store(k1 * m, (v4f *)(out_masked + base1));
    __builtin_nontemporal_store(m4,     (v4f *)(out_mask   + mbase));
}

// ----------------------------- threefry2x32 --------------------------------
__device__ __forceinline__ uint32_t rotl32(uint32_t v, int r) {
    return (v << r) | (v >> (32 - r));
}

// JAX threefry2x32: 20 rounds, key injection every 4 rounds.
__device__ __forceinline__ void threefry2x32(uint32_t k0, uint32_t k1,
                                             uint32_t c0, uint32_t c1,
                                             uint32_t &o0, uint32_t &o1) {
    const uint32_t ks2 = k0 ^ k1 ^ 0x1BD11BDAu;
    uint32_t x0 = c0 + k0, x1 = c1 + k1;
    // rounds 1-4 (13,15,26,6)
    x0 += x1; x1 = rotl32(x1, 13); x1 ^= x0;
    x0 += x1; x1 = rotl32(x1, 15); x1 ^= x0;
    x0 += x1; x1 = rotl32(x1, 26); x1 ^= x0;
    x0 += x1; x1 = rotl32(x1,  6); x1 ^= x0;
    x0 += k1;  x1 += ks2 + 1u;
    // rounds 5-8 (17,29,16,24)
    x0 += x1; x1 = rotl32(x1, 17); x1 ^= x0;
    x0 += x1; x1 = rotl32(x1, 29); x1 ^= x0;
    x0 += x1; x1 = rotl32(x1, 16); x1 ^= x0;
    x0 += x1; x1 = rotl32(x1, 24); x1 ^= x0;
    x0 += ks2; x1 += k0 + 2u;
    // rounds 9-12 (13,15,26,6)
    x0 += x1; x1 = rotl32(x1, 13); x1 ^= x0;
    x0 += x1; x1 = rotl32(x1, 15); x1 ^= x0;
    x0 += x1; x1 = rotl32(x1, 26); x1 ^= x0;
    x0 += x1; x1 = rotl32(x1,  6); x1 ^= x0;
    x0 += k0;  x1 += k1 + 3u;
    // rounds 13-16 (17,29,16,24)
    x0 += x1; x1 = rotl32(x1, 17); x1 ^= x0;
    x0 += x1; x1 = rotl32(x1, 29); x1 ^= x0;
    x0 += x1; x1 = rotl32(x1, 16); x1 ^= x0;
    x0 += x1; x1 = rotl32(x1, 24); x1 ^= x0;
    x0 += k1;  x1 += ks2 + 4u;
    // rounds 17-20 (13,15,26,6)
    x0 += x1; x1 = rotl32(x1, 13); x1 ^= x0;
    x0 += x1; x1 = rotl32(x1, 15); x1 ^= x0;
    x0 += x1; x1 = rotl32(x1, 26); x1 ^= x0;
    x0 += x1; x1 = rotl32(x1,  6); x1 ^= x0;
    x0 += ks2; x1 += k0 + 5u;
    o0 = x0; o1 = x1;
}

// Deterministic ordered block sum for 320 threads (10 wave32 waves).
// sh must hold >= 11 floats.
__device__ float block_sum_320(float v, volatile float *sh) {
    __syncthreads();                       // protect sh reuse across calls
    #pragma unroll
    for (int off = 16; off > 0; off >>= 1)
        v += __shfl_down(v, off, 32);      // wave32 tree reduce, fixed order
    const int wid  = threadIdx.x >> 5;
    const int lane = threadIdx.x & 31;
    if (lane == 0) sh[wid] = v;
    __syncthreads();
    if (threadIdx.x == 0) {
        float s = 0.0f;
        #pragma unroll
        for (int i = 0; i < 10; i++) s += sh[i];   // fixed order => deterministic
        sh[10] = s;
    }
    __syncthreads();
    return sh[10];
}

// ------------------------- stage 1: mask kernel ----------------------------
__global__ __launch_bounds__(NL)
void loupe_mask_kernel(const float *__restrict__ logits,
                       const float *__restrict__ sparsity_p,
                       float *__restrict__ mask_out) {
    __shared__ float sh[11];
    const int tid = threadIdx.x;                 // 0..319
    const float sparsity = *sparsity_p;

    // sigmoid(5*logit)
    float x = 1.0f / (1.0f + expf(-SLOPE * logits[tid]));

    // RescaleProbMap
    const float xbar = block_sum_320(x, sh) * (1.0f / (float)NL);
    const float r    = sparsity / xbar;
    const float beta = (1.0f - sparsity) / (1.0f - xbar);
    x = (r <= 1.0f) ? (x * r) : (1.0f - (1.0f - x) * beta);

    const float xm = block_sum_320(x, sh) * (1.0f / (float)NL);

    // Rejection sampling with JAX threefry PRNG semantics.
    // key0 = jax.random.key(42) = (0, 42)
    uint32_t k0 = 0u, k1 = 42u;
    float result = 0.0f;
    for (int iter = 0; iter < 1024; ++iter) {
        // key, sub = jax.random.split(key): threefry(key, [0,1,2,3]) halves
        uint32_t a0, a1, b0, b1;
        threefry2x32(k0, k1, 0u, 2u, a0, a1);    // block (0,2)
        threefry2x32(k0, k1, 1u, 3u, b0, b1);    // block (1,3)
        // new key = (a0, b0); sub = (a1, b1)

        // uniform(sub, (320,)): counts iota(320) split into halves (i, i+160)
        uint32_t c0, c1, o0, o1;
        if (tid < 160) { c0 = (uint32_t)tid;          c1 = (uint32_t)(tid + 160); }
        else           { c0 = (uint32_t)(tid - 160);  c1 = (uint32_t)tid;         }
        threefry2x32(a1, b1, c0, c1, o0, o1);
        const uint32_t bits = (tid < 160) ? o0 : o1;
        // bits -> float in [0,1): mantissa trick then subtract 1
        const float u = __uint_as_float((bits >> 9) | 0x3F800000u) - 1.0f;

        const float res  = (x > u) ? 1.0f : 0.0f;
        const float mean = block_sum_320(res, sh) * (1.0f / (float)NL);
        result = res;
        // torch.isclose defaults with atol=1e-3: |mean - xm| <= atol + rtol*|xm|
        if (fabsf(mean - xm) <= 1e-3f + 1e-5f * fabsf(xm)) break;  // uniform branch
        k0 = a0; k1 = b0;                                           // consume key
    }

    // preselect outermost lines (PRE_ONE_SIDE = 1)
    if (tid == 0 || tid == NL - 1) result = 1.0f;
    mask_out[tid] = result;
}

// ------------------------------ launcher -----------------------------------
extern "C" void kernel_launch(void *const *d_in, const int *in_sizes, int n_in,
                              void *d_out, int out_size, void *d_ws, size_t ws_size,
                              hipStream_t stream) {
    const float *kspace   = (const float *)d_in[0];  // (256,2,320,320) f32
    const float *sparsity = (const float *)d_in[1];  // scalar f32
    const float *logits   = (const float *)d_in[2];  // (320,) f32

    float *mask       = (float *)d_ws;               // 320 floats of scratch
    float *out_masked = (float *)d_out;              // (256,2,320,320)
    float *out_mask   = out_masked + (size_t)NB * 2 * HW;  // (256,1,320,320)

    loupe_mask_kernel<<<1, NL, 0, stream>>>(logits, sparsity, mask);

    // 80 float4s per row (x), 4 rows per block (y); 80 row-groups x 256 batches
    dim3 block(80, 4, 1);
    dim3 grid(80, NB, 1);
    loupe_apply_kernel<<<grid, block, 0, stream>>>(kspace, mask, out_masked, out_mask);
}